// StockGraphSAGE_2911987826952
// MI455X (gfx1250) — compile-verified
//
#include <hip/hip_runtime.h>

typedef __attribute__((ext_vector_type(2))) float v2f;
typedef __attribute__((ext_vector_type(8))) float v8f;
typedef __attribute__((ext_vector_type(4))) unsigned int v4u;
typedef __attribute__((ext_vector_type(8))) int v8i;
typedef __attribute__((ext_vector_type(4))) int v4i;

#define NN 50000
#define NE 800000

// ---------------- zero scratch (atomic accumulators) ----------------
__global__ void zero_f32(float* __restrict__ p, long n) {
  long i = (long)blockIdx.x * blockDim.x + threadIdx.x;
  long stride = (long)gridDim.x * blockDim.x;
  for (; i < n; i += stride) p[i] = 0.0f;
}

// ---------------- layer-1 aggregation: wave per edge, 30 feats + count ----
__global__ void edge_agg30(const float* __restrict__ x,
                           const long long* __restrict__ src,
                           const long long* __restrict__ dst,
                           float* __restrict__ agg, float* __restrict__ cnt,
                           int n_edges) {
  int e = (int)((blockIdx.x * blockDim.x + threadIdx.x) >> 5);
  int lane = threadIdx.x & 31;
  if (e >= n_edges) return;
  int s = (int)src[e], d = (int)dst[e];
  if (lane < 30) {
    atomicAdd(&agg[(long)d * 30 + lane], x[(long)s * 30 + lane]);
  } else if (lane == 31) {
    atomicAdd(&cnt[d], 1.0f);
  }
}

// ---------------- invcnt = 1/max(cnt,1) ----------------
__global__ void finish_cnt(float* __restrict__ cnt, int n) {
  int i = blockIdx.x * blockDim.x + threadIdx.x;
  if (i < n) cnt[i] = 1.0f / fmaxf(cnt[i], 1.0f);
}

// ---------------- layer-2 aggregation: wave per edge, 128 feats (float4) ---
__global__ void edge_agg128(const float* __restrict__ h,
                            const long long* __restrict__ src,
                            const long long* __restrict__ dst,
                            float* __restrict__ agg, int n_edges) {
  int e = (int)((blockIdx.x * blockDim.x + threadIdx.x) >> 5);
  int lane = threadIdx.x & 31;
  if (e >= n_edges) return;
  int s = (int)src[e], d = (int)dst[e];
  const float4 v = *(const float4*)(h + (long)s * 128 + lane * 4);
  float* ap = agg + (long)d * 128 + lane * 4;
  atomicAdd(ap + 0, v.x);
  atomicAdd(ap + 1, v.y);
  atomicAdd(ap + 2, v.z);
  atomicAdd(ap + 3, v.w);
}

// ---------------- TDM: 1-D copy of n fp32 elements into LDS ----------------
// Builds a Tensor DMA Descriptor (D#) per CDNA5 ISA sec 8.3/8.4:
//   group0: count=1, lds_addr, global_addr[56:0], type=2 ("image")
//   group1: data_size=2 (4B), tensor_dim0=n, tensor_dim1=1,
//           tile_dim0=n (<=65535), tile_dim1=1, stride0=n
// Issued wave-uniform (TDM ignores EXEC); tracked by TENSORcnt.
__device__ __forceinline__ void tdm_load_f32(unsigned lds_addr,
                                             const float* gptr, unsigned n) {
  unsigned long long ga = (unsigned long long)(size_t)gptr;
  v4u g0;
  g0.x = 1u;                                   // count = 1 valid descriptor
  g0.y = lds_addr;                             // LDS byte address
  g0.z = (unsigned)(ga & 0xffffffffu);         // global_addr[31:0]
  g0.w = (unsigned)((ga >> 32) & 0x01ffffffu)  // global_addr[56:32]
         | 0x80000000u;                        // type = 2
  v8i g1;
  g1[0] = (int)0x00020000u;                    // data_size = 2 -> 4 bytes
  g1[1] = (int)((n & 0xffffu) << 16);          // tensor_dim0[15:0]
  g1[2] = (int)(((n >> 16) & 0xffffu)          // tensor_dim0[31:16]
                | (1u << 16));                 // tensor_dim1 = 1
  g1[3] = (int)((n & 0xffffu) << 16);          // tile_dim0 = n
  g1[4] = 1;                                   // tile_dim1 = 1, tile_dim2 = 0
  g1[5] = (int)n;                              // tensor_dim0_stride[31:0]
  g1[6] = 0;
  g1[7] = 0;
  v4i gz4 = (v4i)0;                            // groups 2/3 unused (<=2D)
  v8i gz8 = (v8i)0;
  __builtin_amdgcn_tensor_load_to_lds(g0, g1, gz4, gz4, gz8, 0);
}

// ---------------- fused SAGE dense layer via fp32 WMMA ----------------
// out[m,:] = relu( (agg[m,:]*invcnt[m]) @ Wl + bl + xin[m,:] @ Wr )
// Block = 256 threads (8 waves); each wave owns one 16-row tile, all FOUT
// cols. Weights [Wl|Wr] staged once per block into LDS by the Tensor Data
// Mover; B fragments then come from LDS. K stepped by 4 via
// V_WMMA_F32_16X16X4_F32. K-loop is split into an unchecked body and a
// single checked tail step so the hot loop has no predicated loads.
// Out-of-range rows are address-clamped, not zeroed: WMMA output rows are
// independent, and tail rows are never stored.
template <int FIN, int FOUT, bool RELU>
__global__ __launch_bounds__(256) void sage_gemm(
    const float* __restrict__ agg, const float* __restrict__ invcnt,
    const float* __restrict__ xin, const float* __restrict__ Wl,
    const float* __restrict__ Wr, const float* __restrict__ bl,
    float* __restrict__ out, int n_nodes) {
  extern __shared__ float wlds[];  // 2 * FIN * FOUT floats
  constexpr int NCT = FOUT / 16;   // column tiles per wave
  constexpr int KFULL = FIN & ~3;  // unchecked K iterations
  constexpr unsigned WELEMS = (unsigned)FIN * FOUT;
  static_assert(WELEMS <= 65535u, "tile_dim0 is 16-bit");

  const int wave = threadIdx.x >> 5;
  const int lane = threadIdx.x & 31;

  if (wave == 0) {  // TDM ignores EXEC: issue from exactly one wave
    unsigned lds0 = (unsigned)(size_t)(&wlds[0]);  // addr[31:0] == LDS offset
    tdm_load_f32(lds0, Wl, WELEMS);
    tdm_load_f32(lds0 + WELEMS * 4u, Wr, WELEMS);
    __builtin_amdgcn_s_wait_tensorcnt(0);
  }
  __syncthreads();

  const int m0 = (blockIdx.x * 8 + wave) * 16;
  const int lrow = lane & 15;         // M (A) / N (B) within tile
  const int khalf = (lane >> 4) * 2;  // lanes 16-31 hold K+2, K+3
  const int arow = m0 + lrow;
  const int crow = arow < n_nodes ? arow : 0;  // clamp for addressing only
  const float meanscale = invcnt[crow];
  const long aoff = (long)crow * FIN;

  v8f acc[NCT];
#pragma unroll
  for (int t = 0; t < NCT; ++t) acc[t] = (v8f)0.0f;

#pragma unroll
  for (int phase = 0; phase < 2; ++phase) {
    const float* A = (phase ? xin : agg) + aoff;
    const float rs = phase ? 1.0f : meanscale;
    const float* B = wlds + (unsigned)phase * WELEMS;
#pragma unroll 2
    for (int k0 = 0; k0 < KFULL; k0 += 4) {  // no bounds checks needed
      const int ka = k0 + khalf;
      v2f afrag;
      afrag.x = A[ka] * rs;
      afrag.y = A[ka + 1] * rs;
#pragma unroll
      for (int ct = 0; ct < NCT; ++ct) {
        const int col = ct * 16 + lrow;
        v2f bfrag;
        bfrag.x = B[(unsigned)ka * FOUT + col];
        bfrag.y = B[(unsigned)(ka + 1) * FOUT + col];
        acc[ct] = __builtin_amdgcn_wmma_f32_16x16x4_f32(
            false, afrag, false, bfrag, (short)0, acc[ct], false, false);
      }
    }
    if (FIN % 4 != 0) {  // single checked tail step (layer 1: K=30)
      const int ka = KFULL + khalf;
      v2f afrag;
      afrag.x = (ka < FIN) ? A[ka] * rs : 0.0f;
      afrag.y = (ka + 1 < FIN) ? A[ka + 1] * rs : 0.0f;
#pragma unroll
      for (int ct = 0; ct < NCT; ++ct) {
        const int col = ct * 16 + lrow;
        v2f bfrag;
        bfrag.x = (ka < FIN) ? B[(unsigned)ka * FOUT + col] : 0.0f;
        bfrag.y = (ka + 1 < FIN) ? B[(unsigned)(ka + 1) * FOUT + col] : 0.0f;
        acc[ct] = __builtin_amdgcn_wmma_f32_16x16x4_f32(
            false, afrag, false, bfrag, (short)0, acc[ct], false, false);
      }
    }
  }

  // C layout: VGPR i -> lanes 0-15: M=i, lanes 16-31: M=8+i; N = lane&15
  const int rbase = m0 + (lane >> 4) * 8;
#pragma unroll
  for (int ct = 0; ct < NCT; ++ct) {
    const int col = ct * 16 + lrow;
    const float b = bl[col];
#pragma unroll
    for (int i = 0; i < 8; ++i) {
      const int row = rbase + i;
      if (row < n_nodes) {
        float v = acc[ct][i] + b;
        if (RELU) v = fmaxf(v, 0.0f);
        out[(long)row * FOUT + col] = v;
      }
    }
  }
}

// ---------------- layer 3: project 256 -> 1 before aggregating ----------
__global__ void layer3_dot(const float* __restrict__ h2,
                           const float* __restrict__ Wl3,
                           const float* __restrict__ Wr3,
                           const float* __restrict__ bl3,
                           float* __restrict__ z, float* __restrict__ r,
                           int n_nodes) {
  int node = (int)((blockIdx.x * blockDim.x + threadIdx.x) >> 5);
  int lane = threadIdx.x & 31;
  if (node >= n_nodes) return;
  const float* row = h2 + (long)node * 256;
  float dl = 0.0f, dr = 0.0f;
#pragma unroll
  for (int j = 0; j < 8; ++j) {
    float v = row[lane + 32 * j];
    dl += v * Wl3[lane + 32 * j];
    dr += v * Wr3[lane + 32 * j];
  }
#pragma unroll
  for (int m = 16; m >= 1; m >>= 1) {
    dl += __shfl_xor(dl, m, 32);
    dr += __shfl_xor(dr, m, 32);
  }
  if (lane == 0) {
    z[node] = dl;
    r[node] = dr + bl3[0];
  }
}

// ---------------- aggregate 1 float per edge ----------------
__global__ void edge_agg1(const float* __restrict__ z,
                          const long long* __restrict__ src,
                          const long long* __restrict__ dst,
                          float* __restrict__ outagg, int n_edges) {
  int e = blockIdx.x * blockDim.x + threadIdx.x;
  if (e >= n_edges) return;
  atomicAdd(&outagg[(int)dst[e]], z[(int)src[e]]);
}

__global__ void combine(const float* __restrict__ outagg,
                        const float* __restrict__ invcnt,
                        const float* __restrict__ r, float* __restrict__ out,
                        int n) {
  int i = blockIdx.x * blockDim.x + threadIdx.x;
  if (i < n) out[i] = outagg[i] * invcnt[i] + r[i];
}

extern "C" void kernel_launch(void* const* d_in, const int* in_sizes, int n_in,
                              void* d_out, int out_size, void* d_ws,
                              size_t ws_size, hipStream_t stream) {
  (void)in_sizes; (void)n_in; (void)out_size; (void)ws_size;
  const float* x = (const float*)d_in[0];
  const long long* ei = (const long long*)d_in[1];  // int64 [2, E]
  const float* Wl1 = (const float*)d_in[2];
  const float* Wr1 = (const float*)d_in[3];
  const float* bl1 = (const float*)d_in[4];
  const float* Wl2 = (const float*)d_in[5];
  const float* Wr2 = (const float*)d_in[6];
  const float* bl2 = (const float*)d_in[7];
  const float* Wl3 = (const float*)d_in[8];
  const float* Wr3 = (const float*)d_in[9];
  const float* bl3 = (const float*)d_in[10];
  float* out = (float*)d_out;

  const long long* src = ei;
  const long long* dst = ei + NE;

  // workspace layout: atomic accumulators first (zeroed each call)
  float* ws = (float*)d_ws;
  float* cnt    = ws; ws += NN;            // becomes invcnt
  float* agg1   = ws; ws += (long)NN * 30;
  float* agg2   = ws; ws += (long)NN * 128;
  float* outagg = ws; ws += NN;
  float* h1     = ws; ws += (long)NN * 128;
  float* h2     = ws; ws += (long)NN * 256;
  float* z      = ws; ws += NN;
  float* r      = ws; ws += NN;

  const long n_zero = (long)NN * (1 + 30 + 128 + 1);
  zero_f32<<<2048, 256, 0, stream>>>((float*)d_ws, n_zero);

  // 3125 row-tiles of 16; 8 tiles (128 rows) per block -> 391 blocks
  const int gemm_blocks = (NN / 16 + 7) / 8;

  // layer 1: aggregate x (30-wide) + degree counts
  edge_agg30<<<(NE + 7) / 8, 256, 0, stream>>>(x, src, dst, agg1, cnt, NE);
  finish_cnt<<<(NN + 255) / 256, 256, 0, stream>>>(cnt, NN);
  sage_gemm<30, 128, true>
      <<<gemm_blocks, 256, 2 * 30 * 128 * sizeof(float), stream>>>(
          agg1, cnt, x, Wl1, Wr1, bl1, h1, NN);
  // layer 2: aggregate h1 (128-wide)
  edge_agg128<<<(NE + 7) / 8, 256, 0, stream>>>(h1, src, dst, agg2, NE);
  sage_gemm<128, 256, true>
      <<<gemm_blocks, 256, 2 * 128 * 256 * sizeof(float), stream>>>(
          agg2, cnt, h1, Wl2, Wr2, bl2, h2, NN);
  // layer 3: project to scalar first, then aggregate 1 float/edge
  layer3_dot<<<(NN + 7) / 8, 256, 0, stream>>>(h2, Wl3, Wr3, bl3, z, r, NN);
  edge_agg1<<<(NE + 255) / 256, 256, 0, stream>>>(z, src, dst, outagg, NE);
  combine<<<(NN + 255) / 256, 256, 0, stream>>>(outagg, cnt, r, out, NN);
}